// GNNModel_26096221290524
// MI455X (gfx1250) — compile-verified
//
#include <hip/hip_runtime.h>
#include <hip/hip_bf16.h>
#include <math.h>

// Problem constants (match reference)
#define NB   100      // batch
#define NN   50000    // nodes
#define NE   600000   // edges
#define DD   128      // hidden dim
#define AA   64       // attention dim
#define NREL 401      // 2R+1 relations

typedef __attribute__((ext_vector_type(2))) float v2f;
typedef __attribute__((ext_vector_type(4))) float v4f;
typedef __attribute__((ext_vector_type(8))) float v8f;

// ---------------------------------------------------------------------------
// Zero the aggregation buffer (grid-stride).
// ---------------------------------------------------------------------------
__global__ void k_zero(float* __restrict__ p, int n) {
  int i = blockIdx.x * blockDim.x + threadIdx.x;
  int stride = gridDim.x * blockDim.x;
  for (; i < n; i += stride) p[i] = 0.0f;
}

// ---------------------------------------------------------------------------
// Tiny projections:
//   rows [0, NREL):       projR[row] = rela[row]      @ Wr
//   rows [NREL, NREL+NB): qb[row-N]  = rela[q_rel[b]] @ Wqr + bqr
// One 64-thread block per output row; 128-long dot per thread. Negligible cost.
// ---------------------------------------------------------------------------
__global__ void k_small_proj(const float* __restrict__ rela,
                             const float* __restrict__ Wr,
                             const float* __restrict__ Wqr,
                             const float* __restrict__ bqr,
                             const int*   __restrict__ q_rel,
                             float* __restrict__ projR,
                             float* __restrict__ qb) {
  const int row = blockIdx.x;
  const int a   = threadIdx.x;              // 0..63
  if (row < NREL) {
    const float* r = rela + (long)row * DD;
    float s = 0.0f;
    #pragma unroll 8
    for (int k = 0; k < DD; ++k) s = fmaf(r[k], Wr[k * AA + a], s);
    projR[row * AA + a] = s;
  } else {
    const int b = row - NREL;
    const float* r = rela + (long)q_rel[b] * DD;
    float s = bqr[a];
    #pragma unroll 8
    for (int k = 0; k < DD; ++k) s = fmaf(r[k], Wqr[k * AA + a], s);
    qb[b * AA + a] = s;
  }
}

// ---------------------------------------------------------------------------
// hWs = hidden(50000x128) @ Ws(128x64), f32 WMMA 16x16x4.
// Block = 128 threads = 4 waves; wave w owns output tile (16 rows x cols 16w..).
// A frag: lane holds hidden[m0+lane%16][k0+2h .. +1]  (h = lane/16)  -> float2
// B frag: VGPR j = Ws[k0+2h+j][n0+lane%16]
// C/D: VGPR r -> row m0 + r + 8h, col n0 + lane%16
// ---------------------------------------------------------------------------
__global__ void k_hWs(const float* __restrict__ hidden,
                      const float* __restrict__ Ws,
                      float* __restrict__ hWs) {
  const int lane = threadIdx.x & 31;
  const int wv   = threadIdx.x >> 5;        // 0..3 -> N tile
  const int m0   = blockIdx.x * 16;
  const int n0   = wv * 16;
  const int h    = lane >> 4;               // lane half
  const int l16  = lane & 15;
  const int m    = m0 + l16;
  const int n    = n0 + l16;

  v8f acc = {};
  #pragma unroll 4
  for (int k0 = 0; k0 < DD; k0 += 4) {
    v2f a = *(const v2f*)(hidden + (long)m * DD + k0 + 2 * h);
    v2f b;
    b.x = Ws[(k0 + 2 * h)     * AA + n];
    b.y = Ws[(k0 + 2 * h + 1) * AA + n];
    acc = __builtin_amdgcn_wmma_f32_16x16x4_f32(
        /*neg_a=*/false, a, /*neg_b=*/false, b,
        /*c_mod=*/(short)0, acc, /*reuse_a=*/false, /*reuse_b=*/false);
  }

  const int mr = m0 + 8 * h;
  #pragma unroll
  for (int r = 0; r < 8; ++r)
    hWs[(long)(mr + r) * AA + n] = acc[r];
}

// ---------------------------------------------------------------------------
// Edge kernel (dominant, L2-bandwidth/atomic bound). One wave32 per edge,
// 8 edges per 256-thread block. Vectorized lane ownership:
//   pre:   lane owns a = 2*lane, 2*lane+1   -> b64 gathers (3 tables + wa)
//   msg:   lane owns d = 4*lane..4*lane+3   -> b128 reads of hidden/rela
//   alpha = sigmoid( sum_a relu(pre[a])*wa[a] + ba )
//   agg[obj][d] += alpha * (hidden[sub][d] + rela[rel][d])   (f32 atomics)
// ---------------------------------------------------------------------------
__global__ void k_edges(const int*   __restrict__ edges,
                        const float* __restrict__ hWs,
                        const float* __restrict__ projR,
                        const float* __restrict__ qb,
                        const float* __restrict__ hidden,
                        const float* __restrict__ rela,
                        const float* __restrict__ wa,
                        const float* __restrict__ ba,
                        float* __restrict__ agg) {
  const int lane = threadIdx.x & 31;
  const int e    = blockIdx.x * 8 + (threadIdx.x >> 5);

  const int* ed  = edges + (long)e * 6;
  const int ridx = ed[0];
  const int rel  = ed[2];
  const int sub  = ed[4];
  const int obj  = ed[5];

  // Attention pre-activation: 64 values, 2 per lane (b64 gathers).
  const int a0 = lane * 2;
  const v2f pw = *(const v2f*)(hWs   + sub  * AA + a0);
  const v2f pr = *(const v2f*)(projR + rel  * AA + a0);
  const v2f pq = *(const v2f*)(qb    + ridx * AA + a0);
  const v2f wv = *(const v2f*)(wa + a0);

  float s = fmaxf(pw.x + pr.x + pq.x, 0.0f) * wv.x
          + fmaxf(pw.y + pr.y + pq.y, 0.0f) * wv.y;
  #pragma unroll
  for (int off = 16; off > 0; off >>= 1)
    s += __shfl_xor(s, off, 32);
  s += ba[0];
  const float alpha = 1.0f / (1.0f + __expf(-s));

  // Message + scatter: 128 values, 4 per lane (b128 reads, 4B f32 atomics).
  const int d0 = lane * 4;
  const v4f hv = *(const v4f*)(hidden + (long)sub * DD + d0);
  const v4f rv = *(const v4f*)(rela   + (long)rel * DD + d0);
  float* ao = agg + (long)obj * DD + d0;
  #pragma unroll
  for (int j = 0; j < 4; ++j) {
    const float m = alpha * (hv[j] + rv[j]);
    __hip_atomic_fetch_add(ao + j, m, __ATOMIC_RELAXED,
                           __HIP_MEMORY_SCOPE_AGENT);
  }
}

// ---------------------------------------------------------------------------
// out = agg(50000x128) @ Wh(128x128), f32 WMMA 16x16x4.
// Block = 256 threads = 8 waves; wave w owns cols 16w..16w+15 of a 16-row tile.
// ---------------------------------------------------------------------------
__global__ void k_out(const float* __restrict__ agg,
                      const float* __restrict__ Wh,
                      float* __restrict__ out) {
  const int lane = threadIdx.x & 31;
  const int wv   = threadIdx.x >> 5;        // 0..7 -> N tile
  const int m0   = blockIdx.x * 16;
  const int n0   = wv * 16;
  const int h    = lane >> 4;
  const int l16  = lane & 15;
  const int m    = m0 + l16;
  const int n    = n0 + l16;

  v8f acc = {};
  #pragma unroll 4
  for (int k0 = 0; k0 < DD; k0 += 4) {
    v2f a = *(const v2f*)(agg + (long)m * DD + k0 + 2 * h);
    v2f b;
    b.x = Wh[(k0 + 2 * h)     * DD + n];
    b.y = Wh[(k0 + 2 * h + 1) * DD + n];
    acc = __builtin_amdgcn_wmma_f32_16x16x4_f32(
        false, a, false, b, (short)0, acc, false, false);
  }

  const int mr = m0 + 8 * h;
  #pragma unroll
  for (int r = 0; r < 8; ++r)
    out[(long)(mr + r) * DD + n] = acc[r];
}

// ---------------------------------------------------------------------------
// Launch
// ---------------------------------------------------------------------------
extern "C" void kernel_launch(void* const* d_in, const int* in_sizes, int n_in,
                              void* d_out, int out_size, void* d_ws, size_t ws_size,
                              hipStream_t stream) {
  // setup_inputs order:
  // 0:q_sub 1:q_rel 2:hidden 3:edges 4:n_node 5:rela_embed
  // 6:Ws 7:Wr 8:Wqr 9:bqr 10:wa 11:ba 12:Wh
  const int*   q_rel  = (const int*)  d_in[1];
  const float* hidden = (const float*)d_in[2];
  const int*   edges  = (const int*)  d_in[3];
  const float* rela   = (const float*)d_in[5];
  const float* Ws     = (const float*)d_in[6];
  const float* Wr     = (const float*)d_in[7];
  const float* Wqr    = (const float*)d_in[8];
  const float* bqr    = (const float*)d_in[9];
  const float* wa     = (const float*)d_in[10];
  const float* ba     = (const float*)d_in[11];
  const float* Wh     = (const float*)d_in[12];
  float*       out    = (float*)d_out;

  // Workspace layout (floats): agg | hWs | projR | qb  (~38.5 MB)
  float* ws    = (float*)d_ws;
  float* agg   = ws;
  float* hWs   = agg   + (size_t)NN * DD;
  float* projR = hWs   + (size_t)NN * AA;
  float* qb    = projR + (size_t)NREL * AA;

  k_zero      <<<2048,      256, 0, stream>>>(agg, NN * DD);
  k_small_proj<<<NREL + NB, AA,  0, stream>>>(rela, Wr, Wqr, bqr, q_rel,
                                              projR, qb);
  k_hWs       <<<NN / 16,   128, 0, stream>>>(hidden, Ws, hWs);
  k_edges     <<<NE / 8,    256, 0, stream>>>(edges, hWs, projR, qb,
                                              hidden, rela, wa, ba, agg);
  k_out       <<<NN / 16,   256, 0, stream>>>(agg, Wh, out);
}